// GCRNN_79242146611448
// MI455X (gfx1250) — compile-verified
//
#include <hip/hip_runtime.h>
#include <math.h>

#define ENTITY_NUM 150002
#define USER_NUM   100000
#define COMP_NUM   50000
#define REL_NUM    200
#define DD         128
#define TT         12
#define EE         250000
#define SS         20000
#define BB         1024
#define COMP_BASE  100000   // USER_ID_MAX + 1
#define SLICES     25
#define TPS        125      // company tiles per slice: 25*125*16 = 50000

typedef __attribute__((ext_vector_type(2))) float v2f;
typedef __attribute__((ext_vector_type(8))) float v8f;

__device__ __forceinline__ float atomic_add_f32(float* p, float v) {
  return __hip_atomic_fetch_add(p, v, __ATOMIC_RELAXED, __HIP_MEMORY_SCOPE_AGENT);
}
__device__ __forceinline__ float sigmoidf_(float x) { return 1.0f / (1.0f + __expf(-x)); }

// ---------------------------------------------------------------------------
// Step kernel 1: edge messages m = h[src]*rel_emb[r], scatter-add into msum/cnt
// One thread per (edge, 4-dim group): float4 loads, HW float atomics.
// ---------------------------------------------------------------------------
__global__ void edge_scatter(const float* __restrict__ h,
                             const float* __restrict__ rel_emb,
                             const int* __restrict__ src,
                             const int* __restrict__ dst,
                             const int* __restrict__ rel,
                             float* __restrict__ msum,
                             float* __restrict__ cnt) {
  unsigned tid = blockIdx.x * blockDim.x + threadIdx.x;
  unsigned e = tid >> 5;
  if (e >= EE) return;
  unsigned d4 = (tid & 31u) * 4u;
  int s = src[e], d = dst[e], r = rel[e];
  const float4 hv = *(const float4*)(h + (size_t)s * DD + d4);
  const float4 rv = *(const float4*)(rel_emb + (size_t)r * DD + d4);
  float* o = msum + (size_t)d * DD + d4;
  atomic_add_f32(o + 0, hv.x * rv.x);
  atomic_add_f32(o + 1, hv.y * rv.y);
  atomic_add_f32(o + 2, hv.z * rv.z);
  atomic_add_f32(o + 3, hv.w * rv.w);
  if ((tid & 31u) == 0u) atomic_add_f32(cnt + d, 1.0f);
}

// ---------------------------------------------------------------------------
// Step kernel 2: x = h + msum/max(cnt,1); also re-zero msum/cnt for next step.
// One block (128 threads) per entity row.
// ---------------------------------------------------------------------------
__global__ void aggregate_x(const float* __restrict__ h,
                            float* __restrict__ msum,
                            float* __restrict__ cnt,
                            float* __restrict__ hnext) {
  int row = blockIdx.x, d = threadIdx.x;
  __shared__ float scnt;
  if (d == 0) scnt = cnt[row];
  __syncthreads();
  size_t idx = (size_t)row * DD + d;
  float denom = fmaxf(scnt, 1.0f);
  hnext[idx] = h[idx] + msum[idx] / denom;
  msum[idx] = 0.0f;
  if (d == 0) cnt[row] = 0.0f;
}

// ---------------------------------------------------------------------------
// Step kernel 3: fused LSTM cell via V_WMMA_F32_16X16X4_F32.
// Block = 256 threads = 8 waves; wave `nt` owns D-columns [16*nt, 16*nt+16).
// Each wave keeps 4 accumulators (i,f,g,o gate tiles for the SAME 16x16
// (row, D-col) footprint) so the nonlinearity fuses in-register.
// gates = X@W_ih^T + H@W_hh^T  (W row-major (512,128) => B[k][n] = W[n][k])
// ---------------------------------------------------------------------------
__global__ void __launch_bounds__(256) lstm_gates(
    const float* __restrict__ X, const float* __restrict__ H,
    const float* __restrict__ Cst,
    const float* __restrict__ Wih, const float* __restrict__ Whh,
    const float* __restrict__ bih, const float* __restrict__ bhh,
    const int* __restrict__ seed,
    float* __restrict__ hn, float* __restrict__ cn) {
  const int mt   = blockIdx.x;            // 0..1249  (20000/16)
  const int nt   = threadIdx.x >> 5;      // 0..7     D tile
  const int lane = threadIdx.x & 31;
  const int mrow = lane & 15;
  const int half = lane >> 4;
  const int kh   = half * 2;              // A/B f32 layout: lanes 16-31 hold K+2,K+3
  const int n    = lane & 15;
  const int nb   = nt * 16;

  const int rowA = mt * 16 + mrow;
  const int srA  = seed[rowA];
  const float* xr = X + (size_t)srA * DD;
  const float* hr = H + (size_t)srA * DD;

  v8f acc[4];
#pragma unroll
  for (int g = 0; g < 4; ++g) acc[g] = (v8f){0, 0, 0, 0, 0, 0, 0, 0};

  // X @ W_ih^T
#pragma unroll 4
  for (int k0 = 0; k0 < DD; k0 += 4) {
    v2f a = *(const v2f*)(xr + k0 + kh);
#pragma unroll
    for (int g = 0; g < 4; ++g) {
      const int wrow = g * DD + nb + n;  // gate g, output col nb+n
      v2f b = *(const v2f*)(Wih + (size_t)wrow * DD + k0 + kh);
      acc[g] = __builtin_amdgcn_wmma_f32_16x16x4_f32(false, a, false, b,
                                                     (short)0, acc[g], false, false);
    }
  }
  // H @ W_hh^T  (accumulate into same tiles)
#pragma unroll 4
  for (int k0 = 0; k0 < DD; k0 += 4) {
    v2f a = *(const v2f*)(hr + k0 + kh);
#pragma unroll
    for (int g = 0; g < 4; ++g) {
      const int wrow = g * DD + nb + n;
      v2f b = *(const v2f*)(Whh + (size_t)wrow * DD + k0 + kh);
      acc[g] = __builtin_amdgcn_wmma_f32_16x16x4_f32(false, a, false, b,
                                                     (short)0, acc[g], false, false);
    }
  }

  // C/D layout: vgpr i, halves -> row m = i + 8*half, col n = lane&15
#pragma unroll
  for (int i = 0; i < 8; ++i) {
    const int m   = mt * 16 + i + half * 8;
    const int col = nb + n;
    const int sr  = seed[m];
    float ig = acc[0][i] + bih[col]          + bhh[col];
    float fg = acc[1][i] + bih[DD + col]     + bhh[DD + col];
    float gg = acc[2][i] + bih[2 * DD + col] + bhh[2 * DD + col];
    float og = acc[3][i] + bih[3 * DD + col] + bhh[3 * DD + col];
    float cp = Cst[(size_t)sr * DD + col];
    float cN = sigmoidf_(fg) * cp + sigmoidf_(ig) * tanhf(gg);
    float hN = sigmoidf_(og) * tanhf(cN);
    cn[(size_t)m * DD + col] = cN;
    hn[(size_t)m * DD + col] = hN;
  }
}

// ---------------------------------------------------------------------------
// Step kernel 4: write hn/cn back to the seed rows (avoids WMMA read/write race)
// ---------------------------------------------------------------------------
__global__ void scatter_seed(const int* __restrict__ seed,
                             const float* __restrict__ hn,
                             const float* __restrict__ cn,
                             float* __restrict__ hnext,
                             float* __restrict__ Cst) {
  int b = blockIdx.x, d = threadIdx.x;
  int sr = seed[b];
  hnext[(size_t)sr * DD + d] = hn[(size_t)b * DD + d];
  Cst[(size_t)sr * DD + d]   = cn[(size_t)b * DD + d];
}

// ---------------------------------------------------------------------------
// comp loss: partial online logsumexp of u @ all_c^T via WMMA f32.
// grid = (64 user row-blocks, 25 company slices), block = 1 wave.
// U tile staged in LDS; per-lane running (max,sumexp); cross-lane LSE merge
// over the 16 lanes that own a C row using __shfl_xor.
// ---------------------------------------------------------------------------
__global__ void __launch_bounds__(32) comp_lse_partial(
    const float* __restrict__ h, const float* __restrict__ ent_emb,
    const int* __restrict__ user_batch,
    float* __restrict__ pm, float* __restrict__ ps) {
  const int rb   = blockIdx.x;   // 0..63
  const int sl   = blockIdx.y;   // 0..24
  const int lane = threadIdx.x;
  const int mrow = lane & 15;
  const int half = lane >> 4;
  const int kh   = half * 2;
  const int n    = lane & 15;

  __shared__ float U[16 * DD];
  for (int idx = lane; idx < 16 * DD; idx += 32) {
    int i = idx >> 7, d = idx & 127;
    int user = user_batch[rb * 16 + i];
    U[idx] = h[(size_t)user * DD + d];
  }
  __syncthreads();

  const float* allc = ent_emb + (size_t)COMP_BASE * DD;

  float mr[8], sr[8];
#pragma unroll
  for (int i = 0; i < 8; ++i) { mr[i] = -INFINITY; sr[i] = 0.0f; }

  for (int tt = 0; tt < TPS; ++tt) {
    const int cb = (sl * TPS + tt) * 16;
    v8f acc = (v8f){0, 0, 0, 0, 0, 0, 0, 0};
    const float* crow = allc + (size_t)(cb + n) * DD;
#pragma unroll 8
    for (int k0 = 0; k0 < DD; k0 += 4) {
      v2f a = *(const v2f*)(&U[mrow * DD + k0 + kh]);
      v2f b = *(const v2f*)(crow + k0 + kh);
      acc = __builtin_amdgcn_wmma_f32_16x16x4_f32(false, a, false, b,
                                                  (short)0, acc, false, false);
    }
#pragma unroll
    for (int i = 0; i < 8; ++i) {
      float v  = acc[i];
      float mN = fmaxf(mr[i], v);
      sr[i] = sr[i] * __expf(mr[i] - mN) + __expf(v - mN);
      mr[i] = mN;
    }
  }
  // merge (m,s) across the 16 lanes of each half (xor masks stay in-half)
#pragma unroll
  for (int mask = 1; mask < 16; mask <<= 1) {
#pragma unroll
    for (int i = 0; i < 8; ++i) {
      float om = __shfl_xor(mr[i], mask, 32);
      float os = __shfl_xor(sr[i], mask, 32);
      float mN = fmaxf(mr[i], om);
      sr[i] = sr[i] * __expf(mr[i] - mN) + os * __expf(om - mN);
      mr[i] = mN;
    }
  }
  if (n == 0) {
#pragma unroll
    for (int i = 0; i < 8; ++i) {
      int user = rb * 16 + i + half * 8;
      pm[(size_t)user * SLICES + sl] = mr[i];
      ps[(size_t)user * SLICES + sl] = sr[i];
    }
  }
}

__global__ void comp_finalize(const float* __restrict__ h,
                              const float* __restrict__ ent_emb,
                              const int* __restrict__ user_batch,
                              const int* __restrict__ comp_target,
                              const float* __restrict__ pm,
                              const float* __restrict__ ps,
                              float* __restrict__ out) {
  int b = blockIdx.x * blockDim.x + threadIdx.x;
  if (b >= BB) return;
  float M = -INFINITY, S = 0.0f;
  for (int s = 0; s < SLICES; ++s) {
    float om = pm[(size_t)b * SLICES + s], os = ps[(size_t)b * SLICES + s];
    float mN = fmaxf(M, om);
    S = S * __expf(M - mN) + os * __expf(om - mN);
    M = mN;
  }
  float lse = M + __logf(S);
  int user = user_batch[b];
  const float* u  = h + (size_t)user * DD;
  const float* tc = ent_emb + (size_t)(COMP_BASE + comp_target[b]) * DD;
  float pos = 0.0f;
  for (int k = 0; k < DD; ++k) pos += u[k] * tc[k];
  atomic_add_f32(out + 0, -(pos - lse));
}

// job loss: only 100 relations -> one 128-thread block per user.
__global__ void job_loss(const float* __restrict__ h,
                         const float* __restrict__ rel_emb,
                         const int* __restrict__ user_batch,
                         const int* __restrict__ job_target,
                         float* __restrict__ out) {
  __shared__ float U[DD];
  __shared__ float L[REL_NUM / 2];
  int b = blockIdx.x, d = threadIdx.x;
  int user = user_batch[b];
  U[d] = h[(size_t)user * DD + d];
  __syncthreads();
  if (d < REL_NUM / 2) {
    const float* jr = rel_emb + (size_t)d * DD;
    float dot = 0.0f;
    for (int k = 0; k < DD; ++k) dot += U[k] * jr[k];
    L[d] = dot;
  }
  __syncthreads();
  if (d == 0) {
    float M = -INFINITY;
    for (int j = 0; j < REL_NUM / 2; ++j) M = fmaxf(M, L[j]);
    float S = 0.0f;
    for (int j = 0; j < REL_NUM / 2; ++j) S += __expf(L[j] - M);
    float lse = M + __logf(S);
    const float* tj = rel_emb + (size_t)job_target[b] * DD;
    float pos = 0.0f;
    for (int k = 0; k < DD; ++k) pos += U[k] * tj[k];
    atomic_add_f32(out + 1, -(pos - lse));
  }
}

// ---------------------------------------------------------------------------
extern "C" void kernel_launch(void* const* d_in, const int* in_sizes, int n_in,
                              void* d_out, int out_size, void* d_ws, size_t ws_size,
                              hipStream_t stream) {
  const float* ent_emb = (const float*)d_in[0];
  const float* c0_emb  = (const float*)d_in[1];
  const float* rel_emb = (const float*)d_in[2];
  const float* W_ih    = (const float*)d_in[3];
  const float* W_hh    = (const float*)d_in[4];
  const float* b_ih    = (const float*)d_in[5];
  const float* b_hh    = (const float*)d_in[6];
  const int* src        = (const int*)d_in[7];
  const int* dst        = (const int*)d_in[8];
  const int* rel        = (const int*)d_in[9];
  const int* seed_idx   = (const int*)d_in[10];
  const int* user_batch = (const int*)d_in[11];
  const int* comp_target= (const int*)d_in[12];
  const int* job_target = (const int*)d_in[13];
  float* out = (float*)d_out;

  float* ws = (float*)d_ws;
  const size_t HROW = (size_t)ENTITY_NUM * DD;     // 19,200,256 floats
  float* h0   = ws;
  float* h1   = h0 + HROW;
  float* Cst  = h1 + HROW;
  float* msum = Cst + HROW;
  float* cnt  = msum + HROW;
  float* hnb  = cnt + ((ENTITY_NUM + 63) / 64) * 64;
  float* cnb  = hnb + (size_t)SS * DD;
  float* pm   = cnb + (size_t)SS * DD;
  float* ps   = pm + (size_t)BB * SLICES;

  hipMemcpyAsync(h0, ent_emb, HROW * sizeof(float), hipMemcpyDeviceToDevice, stream);
  hipMemcpyAsync(Cst, c0_emb, HROW * sizeof(float), hipMemcpyDeviceToDevice, stream);
  hipMemsetAsync(msum, 0, HROW * sizeof(float), stream);
  hipMemsetAsync(cnt, 0, ENTITY_NUM * sizeof(float), stream);
  hipMemsetAsync(d_out, 0, 2 * sizeof(float), stream);

  float* hcur = h0;
  float* hnext = h1;
  for (int t = 0; t < TT; ++t) {
    edge_scatter<<<(EE * 32 + 255) / 256, 256, 0, stream>>>(
        hcur, rel_emb, src + (size_t)t * EE, dst + (size_t)t * EE,
        rel + (size_t)t * EE, msum, cnt);
    aggregate_x<<<ENTITY_NUM, DD, 0, stream>>>(hcur, msum, cnt, hnext);
    lstm_gates<<<SS / 16, 256, 0, stream>>>(hnext, hcur, Cst, W_ih, W_hh,
                                            b_ih, b_hh, seed_idx + (size_t)t * SS,
                                            hnb, cnb);
    scatter_seed<<<SS, DD, 0, stream>>>(seed_idx + (size_t)t * SS, hnb, cnb,
                                        hnext, Cst);
    float* tmp = hcur; hcur = hnext; hnext = tmp;
  }

  dim3 g(BB / 16, SLICES);
  comp_lse_partial<<<g, 32, 0, stream>>>(hcur, ent_emb, user_batch, pm, ps);
  comp_finalize<<<(BB + 255) / 256, 256, 0, stream>>>(hcur, ent_emb, user_batch,
                                                      comp_target, pm, ps, out);
  job_loss<<<BB, DD, 0, stream>>>(hcur, rel_emb, user_batch, job_target, out);
}